// PhonologicalLoopMemory_2619930050893
// MI455X (gfx1250) — compile-verified
//
#include <hip/hip_runtime.h>
#include <stdint.h>

#define FEATURE_DIM 128
#define WINDOW_LEN  256
#define BUFFER_LEN  64
#define NUM_RECENT  4
#define BATCH       64
#define FW          (FEATURE_DIM * WINDOW_LEN)   // 32768 floats per slot
#define CHUNKS      4
#define CHUNK_ELEMS (FW / CHUNKS)                // 8192 floats per chunk
#define THREADS     256
#define VEC_PER_THREAD (CHUNK_ELEMS / (THREADS * 4)) // 8 x float4 per thread

// Output layout per batch b (row stride 5*FW):
//   sec 0      : recent[b,0]  == features[b]              (verbatim copy)
//   sec 1..3   : recent[b,j]  == valid ? 0.9*feature_buffer[b,(pos-j)%64] : 0
//   sec 4      : rehearsal    == features[b]              (verbatim copy)
__global__ __launch_bounds__(THREADS)
void plm_kernel(const float* __restrict__ features,
                const float* __restrict__ feature_buffer,
                const int* __restrict__ current_pos,
                const unsigned char* __restrict__ buffer_filled,
                float* __restrict__ out)
{
    // staging for the async DMA copy path: 256 lanes * 8 * 16B = 32 KB
    __shared__ char stage[THREADS * VEC_PER_THREAD * 16];

    int blk = blockIdx.x;
    const int chunk = blk & (CHUNKS - 1); blk >>= 2;
    const int sec   = blk % 5;
    const int b     = blk / 5;
    const int tid   = threadIdx.x;

    const long outBase = (long)b * (5L * FW) + (long)sec * FW + (long)chunk * CHUNK_ELEMS;

    if (sec == 0 || sec == NUM_RECENT) {
        // ---- pure copy of features[b]: async global -> LDS -> global ----
        const char* src  = (const char*)(features + (long)b * FW + (long)chunk * CHUNK_ELEMS);
        char*       dstc = (char*)(out + outBase);
        // low 32 bits of the flat shared pointer == wave-relative LDS byte address
        const uint32_t lds_base = (uint32_t)(uintptr_t)(&stage[0]);

        #pragma unroll
        for (int i = 0; i < VEC_PER_THREAD; ++i) {
            uint64_t ga = (uint64_t)(uintptr_t)(src + ((size_t)i * THREADS + tid) * 16);
            uint32_t la = lds_base + ((uint32_t)i * THREADS + (uint32_t)tid) * 16u;
            asm volatile("global_load_async_to_lds_b128 %0, %1, off"
                         :: "v"(la), "v"(ga) : "memory");
        }
        // loads complete (LDS written) before the dependent async stores issue
        asm volatile("s_wait_asynccnt 0" ::: "memory");
        #pragma unroll
        for (int i = 0; i < VEC_PER_THREAD; ++i) {
            uint64_t ga = (uint64_t)(uintptr_t)(dstc + ((size_t)i * THREADS + tid) * 16);
            uint32_t la = lds_base + ((uint32_t)i * THREADS + (uint32_t)tid) * 16u;
            asm volatile("global_store_async_from_lds_b128 %0, %1, off"
                         :: "v"(ga), "v"(la) : "memory");
        }
        // async stores drain at s_endpgm's implicit wait-idle
    } else {
        // ---- decayed sections: j = 1..3 ----
        const int  j      = sec;
        const int  pos    = current_pos[b];
        const bool filled = buffer_filled[b] != 0;
        const bool valid  = filled || (j <= pos);

        float4* dst = (float4*)(out + outBase);

        if (valid) {
            const int p = (pos - j + BUFFER_LEN) & (BUFFER_LEN - 1);
            const float4* src = (const float4*)(feature_buffer
                                + ((long)b * BUFFER_LEN + p) * FW
                                + (long)chunk * CHUNK_ELEMS);
            #pragma unroll
            for (int i = 0; i < VEC_PER_THREAD; ++i) {
                float4 v = src[i * THREADS + tid];
                v.x *= 0.9f; v.y *= 0.9f; v.z *= 0.9f; v.w *= 0.9f;
                dst[i * THREADS + tid] = v;
            }
        } else {
            const float4 z = make_float4(0.f, 0.f, 0.f, 0.f);
            #pragma unroll
            for (int i = 0; i < VEC_PER_THREAD; ++i)
                dst[i * THREADS + tid] = z;
        }
    }
}

extern "C" void kernel_launch(void* const* d_in, const int* in_sizes, int n_in,
                              void* d_out, int out_size, void* d_ws, size_t ws_size,
                              hipStream_t stream) {
    const float*         features       = (const float*)d_in[0];
    const float*         feature_buffer = (const float*)d_in[1];
    const int*           current_pos    = (const int*)d_in[2];
    const unsigned char* buffer_filled  = (const unsigned char*)d_in[3]; // jax bool: 1 byte
    float*               out            = (float*)d_out;

    dim3 grid(BATCH * 5 * CHUNKS); // 1280 blocks
    plm_kernel<<<grid, THREADS, 0, stream>>>(features, feature_buffer,
                                             current_pos, buffer_filled, out);
}